// GatePQRNNCell_17944373363205
// MI455X (gfx1250) — compile-verified
//
#include <hip/hip_runtime.h>

typedef float v2f __attribute__((ext_vector_type(2)));
typedef float v8f __attribute__((ext_vector_type(8)));

#define DIM 2048

// ---------------------------------------------------------------------------
// Real 16x16x16 matmul accumulate built from four chained V_WMMA_F32_16X16X4_F32.
// A,B are LDS-resident 16x16 row-major f32. asign folds the complex
// subtraction (Dre = AreBre - AimBim) into the A fragment (F32 WMMA has no
// A/B NEG modifier, only CNeg) -> compiler emits v_xor 0x80000000.
// A-frag (16x4): lane m=lane&15; VGPR0=K(koff), VGPR1=K(koff+1), koff=2*(lane>=16)
// B-frag (4x16): lane n=lane&15; VGPR0=row(koff), VGPR1=row(koff+1)
// ---------------------------------------------------------------------------
__device__ __forceinline__ v8f wmma_acc(const float* __restrict__ A,
                                        const float* __restrict__ B,
                                        v8f acc, float asign, int mn, int koff) {
#pragma unroll
  for (int k0 = 0; k0 < 16; k0 += 4) {
    v2f a, b;
    a[0] = asign * A[mn * 16 + k0 + koff];
    a[1] = asign * A[mn * 16 + k0 + koff + 1];
    b[0] = B[(k0 + koff) * 16 + mn];
    b[1] = B[(k0 + koff + 1) * 16 + mn];
    acc = __builtin_amdgcn_wmma_f32_16x16x4_f32(false, a, false, b, (short)0,
                                                acc, false, false);
  }
  return acc;
}

// Complex 16x16 @ 16x16 tile product: D = A * B (A,B complex, LDS resident).
__device__ __forceinline__ void cmul_tiles(const float* Are, const float* Aim,
                                           const float* Bre, const float* Bim,
                                           int lane, v8f& Dre, v8f& Dim) {
  int mn = lane & 15;
  int koff = (lane >> 4) << 1;
  v8f zero = {};
  v8f pre = wmma_acc(Are, Bre, zero, 1.0f, mn, koff);
  pre = wmma_acc(Aim, Bim, pre, -1.0f, mn, koff);
  v8f pim = wmma_acc(Are, Bim, zero, 1.0f, mn, koff);
  pim = wmma_acc(Aim, Bre, pim, 1.0f, mn, koff);
  Dre = pre;
  Dim = pim;
}

// ---------------------------------------------------------------------------
// 2x2 complex matrix multiply helper (gate composition)
// ---------------------------------------------------------------------------
__device__ __forceinline__ void c2mul(const float ar[2][2], const float ai[2][2],
                                      const float br[2][2], const float bi[2][2],
                                      float cr[2][2], float ci[2][2]) {
#pragma unroll
  for (int i = 0; i < 2; ++i)
#pragma unroll
    for (int j = 0; j < 2; ++j) {
      float re = 0.f, im = 0.f;
#pragma unroll
      for (int k = 0; k < 2; ++k) {
        re += ar[i][k] * br[k][j] - ai[i][k] * bi[k][j];
        im += ar[i][k] * bi[k][j] + ai[i][k] * br[k][j];
      }
      cr[i][j] = re;
      ci[i][j] = im;
    }
}

// ---------------------------------------------------------------------------
// prep: per-wire composed 2x2 unitaries -> Kronecker groups + ZZ phase vector
// ws layout (floats): [0:256) WhiRe [256:512) WhiIm [512:768) WmidRe
// [768:1024) WmidIm [1024:1280) Wlo16Re [1280:1536) Wlo16Im
// [1536:5632) p[2048] interleaved re/im. rho at float offset 8192.
// ---------------------------------------------------------------------------
__global__ void __launch_bounds__(256)
prep_kernel(const float* __restrict__ x, const float* __restrict__ rot,
            const float* __restrict__ zzp, float* __restrict__ ws) {
  __shared__ float Vre[11][2][2], Vim[11][2][2];
  int tid = threadIdx.x;
  if (tid < 11) {
    int w = tid;
    float r0 = rot[w * 3 + 0], r1 = rot[w * 3 + 1], r2 = rot[w * 3 + 2];
    float c0 = cosf(0.5f * r0), s0 = sinf(0.5f * r0);
    float Ar[2][2] = {{c0, 0.f}, {0.f, c0}};
    float Ai[2][2] = {{0.f, -s0}, {-s0, 0.f}};        // RX(r0)
    float c1 = cosf(0.5f * r1), s1 = sinf(0.5f * r1);
    float Zr[2][2] = {{c1, 0.f}, {0.f, c1}};
    float Zi[2][2] = {{-s1, 0.f}, {0.f, s1}};         // RZ(r1)
    float c2 = cosf(0.5f * r2), s2 = sinf(0.5f * r2);
    float Br[2][2] = {{c2, 0.f}, {0.f, c2}};
    float Bi[2][2] = {{0.f, -s2}, {-s2, 0.f}};        // RX(r2)
    float T1r[2][2], T1i[2][2], T2r[2][2], T2i[2][2];
    c2mul(Zr, Zi, Ar, Ai, T1r, T1i);                  // RZ*RX0
    c2mul(Br, Bi, T1r, T1i, T2r, T2i);                // RX2*RZ*RX0
    if (w < 6) {                                      // data wires: * RY(x_w)
      float cy = cosf(0.5f * x[w]), sy = sinf(0.5f * x[w]);
      float Yr[2][2] = {{cy, -sy}, {sy, cy}};
      float Yi[2][2] = {{0.f, 0.f}, {0.f, 0.f}};
      float T3r[2][2], T3i[2][2];
      c2mul(T2r, T2i, Yr, Yi, T3r, T3i);
#pragma unroll
      for (int i = 0; i < 2; ++i)
#pragma unroll
        for (int j = 0; j < 2; ++j) { Vre[w][i][j] = T3r[i][j]; Vim[w][i][j] = T3i[i][j]; }
    } else {
#pragma unroll
      for (int i = 0; i < 2; ++i)
#pragma unroll
        for (int j = 0; j < 2; ++j) { Vre[w][i][j] = T2r[i][j]; Vim[w][i][j] = T2i[i][j]; }
    }
  }
  __syncthreads();
  int a = tid >> 4, b = tid & 15;
  {  // Whi = V0 (x) V1 (x) V2 (x) V3
    float re = 1.f, im = 0.f;
#pragma unroll
    for (int w = 0; w < 4; ++w) {
      int aw = (a >> (3 - w)) & 1, bw = (b >> (3 - w)) & 1;
      float vr = Vre[w][aw][bw], vi = Vim[w][aw][bw];
      float nr = re * vr - im * vi, ni = re * vi + im * vr;
      re = nr; im = ni;
    }
    ws[tid] = re;
    ws[256 + tid] = im;
  }
  {  // Wmid = V4 (x) ... (x) V7
    float re = 1.f, im = 0.f;
#pragma unroll
    for (int w = 0; w < 4; ++w) {
      int aw = (a >> (3 - w)) & 1, bw = (b >> (3 - w)) & 1;
      float vr = Vre[4 + w][aw][bw], vi = Vim[4 + w][aw][bw];
      float nr = re * vr - im * vi, ni = re * vi + im * vr;
      re = nr; im = ni;
    }
    ws[512 + tid] = re;
    ws[768 + tid] = im;
  }
  {  // Wlo16 = blockdiag(V8(x)V9(x)V10, same): one 16x16 WMMA applies both 8-groups
    float re = 0.f, im = 0.f;
    if ((a >> 3) == (b >> 3)) {
      int a3 = a & 7, b3 = b & 7;
      re = 1.f;
#pragma unroll
      for (int w = 0; w < 3; ++w) {
        int aw = (a3 >> (2 - w)) & 1, bw = (b3 >> (2 - w)) & 1;
        float vr = Vre[8 + w][aw][bw], vi = Vim[8 + w][aw][bw];
        float nr = re * vr - im * vi, ni = re * vi + im * vr;
        re = nr; im = ni;
      }
    }
    ws[1024 + tid] = re;
    ws[1280 + tid] = im;
  }
  // ZZ phases: p[r] = exp(-i/2 * sum_e theta_e * z_a(r) z_b(r)), wire w = bit (10-w)
  for (int r = tid; r < 2048; r += 256) {
    float phi = 0.f;
#pragma unroll
    for (int e = 0; e < 10; ++e) {
      int za = 1 - 2 * ((r >> (10 - e)) & 1);
      int zb = 1 - 2 * ((r >> (9 - e)) & 1);
      phi += -0.5f * zzp[e] * (float)(za * zb);
    }
    ws[1536 + 2 * r] = cosf(phi);
    ws[1536 + 2 * r + 1] = sinf(phi);
  }
}

// ---------------------------------------------------------------------------
// Left sweep: rho[rowbase + t*S, c0..c0+15] <- W * tile, S = 1<<sh.
// One wave per 16x16 tile; tile holds the full contraction axis -> in-place.
// init_mode!=0 (first sweep only): synthesize rho0 = |0><0|_D (x) rho_H on the
// fly instead of reading rho (saves the init write pass + this sweep's reads;
// also guarantees rho is never read-before-write across graph replays).
// ---------------------------------------------------------------------------
__global__ void __launch_bounds__(128)
apply_left_kernel(float2* __restrict__ rho, const float* __restrict__ Wre,
                  const float* __restrict__ Wim, int sh,
                  const float* __restrict__ hre, const float* __restrict__ him,
                  int init_mode) {
  __shared__ float sA[4][2][256];
  __shared__ float sB[4][2][256];
  int wave = threadIdx.x >> 5;
  int lane = threadIdx.x & 31;
  int tile = blockIdx.x * 4 + wave;
  int g = tile >> 7;                       // row group 0..127
  int ct = tile & 127;                     // column tile
  int stride = 1 << sh;
  int lo = g & (stride - 1);
  int hi = g >> sh;
  int rowbase = (hi << (sh + 4)) + lo;
  int c0 = ct << 4;
  float* Are = sA[wave][0]; float* Aim = sA[wave][1];
  float* Bre = sB[wave][0]; float* Bim = sB[wave][1];
#pragma unroll
  for (int q = 0; q < 8; ++q) {
    int e = lane * 8 + q;
    Are[e] = Wre[e];
    Aim[e] = Wim[e];
    int t = e >> 4, n = e & 15;
    int row = rowbase + (t << sh);
    int col = c0 + n;
    float vre, vim;
    if (init_mode) {
      vre = 0.f; vim = 0.f;
      if (row < 32 && col < 32) {          // top-left 32x32 block = rho_H
        vre = hre[row * 32 + col];
        vim = him[row * 32 + col];
      }
    } else {
      float2 v = rho[(size_t)row * DIM + col];
      vre = v.x; vim = v.y;
    }
    Bre[e] = vre;
    Bim[e] = vim;
  }
  __syncthreads();
  v8f Dre, Dim;
  cmul_tiles(Are, Aim, Bre, Bim, lane, Dre, Dim);
  int hb = (lane >> 4) << 3;
  int n = lane & 15;
#pragma unroll
  for (int v = 0; v < 8; ++v) {
    int t = v + hb;   // D layout: VGPR v -> row v + 8*(lane>=16)
    rho[(size_t)(rowbase + (t << sh)) * DIM + c0 + n] = make_float2(Dre[v], Dim[v]);
  }
}

// ---------------------------------------------------------------------------
// Right sweep: rho[r0..r0+15, colbase + k*S] <- tile * W^H
// zz_mode!=0 (last sweep only): fuse the diagonal IsingZZ mask into the store:
// rho[r][c] *= p[r] * conj(p[c])  (each element written exactly once/sweep).
// ---------------------------------------------------------------------------
__global__ void __launch_bounds__(128)
apply_right_kernel(float2* __restrict__ rho, const float* __restrict__ Wre,
                   const float* __restrict__ Wim, int sh,
                   const float* __restrict__ p, int zz_mode) {
  __shared__ float sA[4][2][256];
  __shared__ float sB[4][2][256];
  int wave = threadIdx.x >> 5;
  int lane = threadIdx.x & 31;
  int tile = blockIdx.x * 4 + wave;
  int rt = tile >> 7;                      // 16 consecutive rows
  int g = tile & 127;                      // column group
  int stride = 1 << sh;
  int lo = g & (stride - 1);
  int hi = g >> sh;
  int colbase = (hi << (sh + 4)) + lo;
  int r0 = rt << 4;
  float* Are = sA[wave][0]; float* Aim = sA[wave][1];
  float* Bre = sB[wave][0]; float* Bim = sB[wave][1];
#pragma unroll
  for (int q = 0; q < 8; ++q) {
    int e = lane * 8 + q;
    int k = e >> 4, n = e & 15;
    Bre[e] = Wre[n * 16 + k];              // B = W^H
    Bim[e] = -Wim[n * 16 + k];
    float2 v = rho[(size_t)(r0 + k) * DIM + colbase + (n << sh)];  // A[m][k2]
    Are[e] = v.x;
    Aim[e] = v.y;
  }
  __syncthreads();
  v8f Dre, Dim;
  cmul_tiles(Are, Aim, Bre, Bim, lane, Dre, Dim);
  int hb = (lane >> 4) << 3;
  int n = lane & 15;
  int col = colbase + (n << sh);
  float pcr = 1.f, pci = 0.f;
  if (zz_mode) { pcr = p[2 * col]; pci = p[2 * col + 1]; }
#pragma unroll
  for (int v = 0; v < 8; ++v) {
    int m = v + hb;
    int row = r0 + m;
    float dre = Dre[v], dim = Dim[v];
    if (zz_mode) {
      // f = p[row] * conj(p[col]); d *= f
      float prr = p[2 * row], pri = p[2 * row + 1];
      float fr = prr * pcr + pri * pci;
      float fi = pri * pcr - prr * pci;
      float nre = dre * fr - dim * fi;
      float nim = dre * fi + dim * fr;
      dre = nre; dim = nim;
    }
    rho[(size_t)row * DIM + col] = make_float2(dre, dim);
  }
}

// <Z0> from diagonal (bit 10), then y = ((1-<Z0>)/2)*w + b -> out[0]
__global__ void __launch_bounds__(256)
readout_kernel(const float2* __restrict__ rho, const float* __restrict__ w,
               const float* __restrict__ bb, float* __restrict__ out) {
  __shared__ float red[256];
  int tid = threadIdx.x;
  float s = 0.f;
  for (int r = tid; r < 2048; r += 256) {
    float d = rho[(size_t)r * DIM + r].x;
    s += (r < 1024) ? d : -d;
  }
  red[tid] = s;
  __syncthreads();
  for (int k = 128; k > 0; k >>= 1) {
    if (tid < k) red[tid] += red[tid + k];
    __syncthreads();
  }
  if (tid == 0) {
    float expz = red[0];
    float p1 = 0.5f * (1.f - expz);
    out[0] = p1 * w[0] + bb[0];
  }
}

// Partial trace over data wires: rho_h[b][d] = sum_a rho[a*32+b][a*32+d]
__global__ void __launch_bounds__(256)
ptrace_kernel(const float2* __restrict__ rho, float* __restrict__ out) {
  int idx = blockIdx.x * 256 + threadIdx.x;  // 0..1023
  int b = idx >> 5, d = idx & 31;
  float re = 0.f, im = 0.f;
  for (int a = 0; a < 64; ++a) {
    float2 v = rho[(size_t)(a * 32 + b) * DIM + a * 32 + d];
    re += v.x;
    im += v.y;
  }
  out[1 + 2 * idx] = re;
  out[2 + 2 * idx] = im;
}

extern "C" void kernel_launch(void* const* d_in, const int* in_sizes, int n_in,
                              void* d_out, int out_size, void* d_ws, size_t ws_size,
                              hipStream_t stream) {
  const float* x   = (const float*)d_in[0];
  const float* hre = (const float*)d_in[1];
  const float* him = (const float*)d_in[2];
  const float* rot = (const float*)d_in[3];
  const float* zzp = (const float*)d_in[4];
  const float* rw  = (const float*)d_in[5];
  const float* rb  = (const float*)d_in[6];

  float* wsf = (float*)d_ws;
  float* Whi_re  = wsf;          float* Whi_im  = wsf + 256;
  float* Wmid_re = wsf + 512;    float* Wmid_im = wsf + 768;
  float* Wlo_re  = wsf + 1024;   float* Wlo_im  = wsf + 1280;
  float* p       = wsf + 1536;
  float2* rho    = (float2*)(wsf + 8192);   // 2048x2048 complex64, 33.5 MB

  prep_kernel<<<1, 256, 0, stream>>>(x, rot, zzp, wsf);
  // left (row) sweeps: strides 128, 8, 1 (wires 0-3, 4-7, 8-10);
  // first sweep synthesizes rho0 in place of reading rho.
  apply_left_kernel<<<4096, 128, 0, stream>>>(rho, Whi_re, Whi_im, 7, hre, him, 1);
  apply_left_kernel<<<4096, 128, 0, stream>>>(rho, Wmid_re, Wmid_im, 3, hre, him, 0);
  apply_left_kernel<<<4096, 128, 0, stream>>>(rho, Wlo_re, Wlo_im, 0, hre, him, 0);
  // right (column) sweeps with W^H; last sweep fuses the IsingZZ phase mask.
  apply_right_kernel<<<4096, 128, 0, stream>>>(rho, Whi_re, Whi_im, 7, p, 0);
  apply_right_kernel<<<4096, 128, 0, stream>>>(rho, Wmid_re, Wmid_im, 3, p, 0);
  apply_right_kernel<<<4096, 128, 0, stream>>>(rho, Wlo_re, Wlo_im, 0, p, 1);
  readout_kernel<<<1, 256, 0, stream>>>(rho, rw, rb, (float*)d_out);
  ptrace_kernel<<<4, 256, 0, stream>>>(rho, (float*)d_out);

  (void)in_sizes; (void)n_in; (void)out_size; (void)ws_size;
}